// MemoryEfficientGaussianRasterizer_76897094467844
// MI455X (gfx1250) — compile-verified
//
#include <hip/hip_runtime.h>
#include <hip/hip_bf16.h>
#include <math.h>

typedef __attribute__((ext_vector_type(2))) float v2f;
typedef __attribute__((ext_vector_type(8))) float v8f;

#define ALPHA_TH (1.0f / 255.0f)
#define MAX_ALPHA 0.99f
#define EPSV 1e-08f
#define MAXG 256

// ---------------------------------------------------------------------------
// Kernel 1: stable depth rank + per-gaussian feature precompute, scattered
// into depth order.
//   feat[g][8] = [a, 2b, c, u, v, w0, 0, 0]   (q = feat . [x^2, xy, y^2, x, y, 1])
//   scal[g][8] = [opa, tau, valid, colR, colG, colB, 0, 0]
// ---------------------------------------------------------------------------
__global__ __launch_bounds__(256) void gs_preprocess(
    const float* __restrict__ means2d, const float* __restrict__ conics,
    const float* __restrict__ colors, const float* __restrict__ opacities,
    const float* __restrict__ depths, float* __restrict__ feat,
    float* __restrict__ scal, int G) {
  __shared__ float sdepth[MAXG];
  for (int i = threadIdx.x; i < G; i += blockDim.x) sdepth[i] = depths[i];
  __syncthreads();

  for (int g = threadIdx.x; g < G; g += blockDim.x) {
    float dg = sdepth[g];
    int rank = 0;
    for (int j = 0; j < G; ++j) {
      float dj = sdepth[j];
      rank += (dj < dg) || ((dj == dg) && (j < g)) ? 1 : 0;
    }
    float mx = means2d[2 * g + 0], my = means2d[2 * g + 1];
    float a = conics[3 * g + 0], b = conics[3 * g + 1], c = conics[3 * g + 2];
    float opa = opacities[g];
    float det = a * c - b * b;
    float tau = -2.0f * logf(fmaxf(ALPHA_TH / fmaxf(opa, EPSV), EPSV));
    float valid = ((opa > ALPHA_TH) && (det > EPSV) && (a > 0.0f) &&
                   (c > 0.0f) && (tau > 0.0f))
                      ? 1.0f
                      : 0.0f;
    float u = -2.0f * (a * mx + b * my);
    float v = -2.0f * (b * mx + c * my);
    float w0 = a * mx * mx + 2.0f * b * mx * my + c * my * my;

    float* f = feat + (size_t)rank * 8;
    f[0] = a; f[1] = 2.0f * b; f[2] = c; f[3] = u; f[4] = v; f[5] = w0;
    f[6] = 0.0f; f[7] = 0.0f;
    float* s = scal + (size_t)rank * 8;
    s[0] = opa; s[1] = tau; s[2] = valid;
    s[3] = colors[3 * g + 0]; s[4] = colors[3 * g + 1]; s[5] = colors[3 * g + 2];
    s[6] = 0.0f; s[7] = 0.0f;
  }
}

// ---------------------------------------------------------------------------
// Kernel 2: WMMA rasterizer. Each wave32 owns a 16x1 pixel strip.
// Per chunk: A = 16 gaussian feature rows (rows 0-7 from the low-depth stream,
// rows 8-15 from the high-depth stream), B = pixel features, two
// v_wmma_f32_16x16x4_f32 give q for 16 gaussians x 16 pixels.
// D layout => lanes 0-15 composite stream-lo (8 gaussians/chunk, in order),
// lanes 16-31 composite stream-hi; one shfl-merge at the end preserves the
// global front-to-back order.
// ---------------------------------------------------------------------------
__global__ __launch_bounds__(256) void gs_raster(
    const float* __restrict__ feat_g, const float* __restrict__ scal_g,
    const float* __restrict__ background, float* __restrict__ out, int Wd,
    int Hd, int G) {
  __shared__ float sfeat[MAXG * 8];
  __shared__ float sscal[MAXG * 8];

  // stage gaussian tables into LDS (both arrays together: 16KB)
  {
    const float4* fg4 = (const float4*)feat_g;
    const float4* sg4 = (const float4*)scal_g;
    float4* sf4 = (float4*)sfeat;
    float4* ss4 = (float4*)sscal;
    int n4 = (G * 8) / 4;
    for (int i = threadIdx.x; i < n4; i += 256) {
      sf4[i] = fg4[i];
      ss4[i] = sg4[i];
    }
  }
  __syncthreads();

  const int tid = threadIdx.x;
  const int wave = tid >> 5;
  const int lane = tid & 31;
  const int l = lane & 15;   // pixel column / matrix row-within-half
  const int half = lane >> 4;

  const int y = blockIdx.y;
  const int x = blockIdx.x * 128 + wave * 16 + l;
  const float px = (float)x + 0.5f;
  const float py = (float)y + 0.5f;

  // B operand (pixel features), K split across half-waves:
  // lanes 0-15 hold K=0,1 ; lanes 16-31 hold K=2,3 (per 32-bit 16x4 layout)
  v2f b1, b2;
  b1.x = half ? (py * py) : (px * px);  // K2 : K0
  b1.y = half ? px : (px * py);         // K3 : K1
  b2.x = half ? 0.0f : py;              // K6 : K4
  b2.y = half ? 0.0f : 1.0f;            // K7 : K5

  const int Ghalf = G >> 1;  // split sorted list: lo stream / hi stream
  const int nch = G >> 4;    // 16 gaussians per chunk (8 per stream)

  float accR = 0.0f, accG = 0.0f, accB = 0.0f;  // this half's stream state
  float trans = 1.0f;

  for (int ch = 0; ch < nch; ++ch) {
    // A operand: M row = l ; rows 0-7 -> lo stream, rows 8-15 -> hi stream
    int j = (l < 8) ? (ch * 8 + l) : (Ghalf + ch * 8 + (l - 8));
    const float* fj = sfeat + j * 8;
    v2f a1, a2;
    a1.x = fj[half ? 2 : 0];
    a1.y = fj[half ? 3 : 1];
    a2.x = half ? 0.0f : fj[4];
    a2.y = half ? 0.0f : fj[5];

    v8f q8 = {};
    q8 = __builtin_amdgcn_wmma_f32_16x16x4_f32(false, a1, false, b1, (short)0,
                                               q8, false, false);
    q8 = __builtin_amdgcn_wmma_f32_16x16x4_f32(false, a2, false, b2, (short)0,
                                               q8, false, false);

    // composite 8 gaussians of this half's stream, strictly in depth order
#pragma unroll
    for (int r = 0; r < 8; ++r) {
      int gi = (half ? Ghalf : 0) + ch * 8 + r;  // D: M=r (lo) / M=r+8 (hi)
      const float* s = sscal + gi * 8;
      float opa = s[0], tau = s[1], val = s[2];
      float q = q8[r];
      float al = opa * __expf(-0.5f * q);
      al = fminf(al, MAX_ALPHA);
      al = (q <= tau) ? al : 0.0f;
      al *= val;
      float w = trans * al;
      accR += w * s[3];
      accG += w * s[4];
      accB += w * s[5];
      trans *= (1.0f - al);
    }
  }

  // merge ordered streams: lo (lanes 0-15) first, then hi (lanes 16-31)
  float loR = __shfl(accR, l, 32), hiR = __shfl(accR, l + 16, 32);
  float loG = __shfl(accG, l, 32), hiG = __shfl(accG, l + 16, 32);
  float loB = __shfl(accB, l, 32), hiB = __shfl(accB, l + 16, 32);
  float loT = __shfl(trans, l, 32), hiT = __shfl(trans, l + 16, 32);

  float oR = loR + loT * hiR;
  float oG = loG + loT * hiG;
  float oB = loB + loT * hiB;
  float oT = loT * hiT;

  if (half == 0 && x < Wd && y < Hd) {
    float* o = out + ((size_t)y * Wd + x) * 3;
    o[0] = oR + oT * background[0];
    o[1] = oG + oT * background[1];
    o[2] = oB + oT * background[2];
  }
}

extern "C" void kernel_launch(void* const* d_in, const int* in_sizes, int n_in,
                              void* d_out, int out_size, void* d_ws,
                              size_t ws_size, hipStream_t stream) {
  const float* means2d = (const float*)d_in[0];
  const float* conics = (const float*)d_in[1];
  const float* colors = (const float*)d_in[2];
  const float* opacities = (const float*)d_in[3];
  const float* depths = (const float*)d_in[4];
  const float* background = (const float*)d_in[5];
  const int G = in_sizes[3];  // opacities: (G,)
  const int Wd = 256, Hd = 256;  // reference fixes H=W=256 (out_size = H*W*3)

  float* feat = (float*)d_ws;              // [G][8]
  float* scal = feat + (size_t)G * 8;      // [G][8]
  float* out = (float*)d_out;

  gs_preprocess<<<1, 256, 0, stream>>>(means2d, conics, colors, opacities,
                                       depths, feat, scal, G);
  dim3 grid((Wd + 127) / 128, Hd);
  gs_raster<<<grid, 256, 0, stream>>>(feat, scal, background, out, Wd, Hd, G);
}